// GATV2_ENC_9775345566175
// MI455X (gfx1250) — compile-verified
//
#include <hip/hip_runtime.h>
#include <hip/hip_bf16.h>

typedef __attribute__((ext_vector_type(2))) float v2f;
typedef __attribute__((ext_vector_type(8))) float v8f;

#define BSZ      8
#define TDIM     100
#define NNODE    2048
#define NTOT     (BSZ * NNODE)          /* 16384 */
#define CIN      128
#define HC1      256                    /* H*hid layer 1 */
#define COUT     128
#define NEDGE    32768
#define E_TILED  (NEDGE * BSZ)          /* 262144 */
#define E_TOT    (E_TILED + NTOT)       /* 278528 */
#define TPB      256

// ---------------------------------------------------------------- helpers

__device__ __forceinline__ void edge_nodes(int e, const int* __restrict__ ei,
                                           int& s, int& d) {
  if (e < E_TILED) {
    int i = e & (NEDGE - 1);
    s = ei[i];
    d = ei[NEDGE + i];
  } else {
    s = d = e - E_TILED;   // self loop over all N nodes
  }
}

// monotonic float <-> uint key (for atomicMax); key 0 == minimum
__device__ __forceinline__ unsigned fkey(float f) {
  unsigned u = __float_as_uint(f);
  return (u & 0x80000000u) ? ~u : (u | 0x80000000u);
}
__device__ __forceinline__ float fdec(unsigned k) {
  return __uint_as_float((k & 0x80000000u) ? (k ^ 0x80000000u) : ~k);
}

// ---------------------------------------------------------------- zero fill

__global__ void zero_kernel(float* __restrict__ p, long n) {
  long t = blockIdx.x * (long)blockDim.x + threadIdx.x;
  if (t < n) p[t] = 0.0f;
}

// ---------------------------------------------------------------- temporal embedding GEMM
// x[r, c] = sum_t input[b, t, n] * W[t, c] + bias[c],  r = b*2048 + n
// M=16384, K=100, N=128 ; one wave -> one 16x16 tile via V_WMMA_F32_16X16X4_F32

__global__ void temb_gemm_kernel(const float* __restrict__ input,
                                 const float* __restrict__ W,
                                 const float* __restrict__ bias,
                                 float* __restrict__ x) {
  int wave = (blockIdx.x * blockDim.x + threadIdx.x) >> 5;
  int lane = threadIdx.x & 31;
  int tn = wave & 7;          // 128/16 = 8 col tiles
  int tm = wave >> 3;
  int row = lane & 15;
  int kh  = (lane >> 4) << 1; // 0 or 2
  int r   = tm * 16 + row;
  int bb  = r >> 11;
  int nn  = r & (NNODE - 1);
  const float* ain  = input + (long)bb * TDIM * NNODE + nn;   // + t*NNODE
  const float* bcol = W + tn * 16 + row;                      // col = row lanes
  v8f acc = {};
  for (int k = 0; k < TDIM; k += 4) {
    v2f a, b;
    a.x = ain[(long)(k + kh) * NNODE];
    a.y = ain[(long)(k + kh + 1) * NNODE];
    b.x = bcol[(long)(k + kh) * CIN];
    b.y = bcol[(long)(k + kh + 1) * CIN];
    acc = __builtin_amdgcn_wmma_f32_16x16x4_f32(false, a, false, b,
                                                (short)0, acc, false, false);
  }
  int col = tn * 16 + row;
  float bc = bias[col];
  float* cp = x + (long)(tm * 16 + kh * 4) * CIN + col;  // rows j and j+8
#pragma unroll
  for (int j = 0; j < 8; ++j) cp[(long)j * CIN] = acc[j] + bc;
}

// ---------------------------------------------------------------- generic f32 WMMA GEMM
// C[M,N] = A[M,K] @ B[K,N], row-major, all dims multiples of 16 (K of 4)

template <int KDIM, int NCOLS>
__global__ void wmma_gemm_f32(const float* __restrict__ A,
                              const float* __restrict__ B,
                              float* __restrict__ C) {
  int wave = (blockIdx.x * blockDim.x + threadIdx.x) >> 5;
  int lane = threadIdx.x & 31;
  const int tilesN = NCOLS / 16;
  int tm = wave / tilesN;
  int tn = wave % tilesN;
  int row = lane & 15;
  int kh  = (lane >> 4) << 1;
  const float* arow = A + (long)(tm * 16 + row) * KDIM;
  const float* bcol = B + tn * 16 + row;
  v8f acc = {};
  for (int k = 0; k < KDIM; k += 4) {
    v2f a, b;
    a.x = arow[k + kh];
    a.y = arow[k + kh + 1];
    b.x = bcol[(long)(k + kh) * NCOLS];
    b.y = bcol[(long)(k + kh + 1) * NCOLS];
    acc = __builtin_amdgcn_wmma_f32_16x16x4_f32(false, a, false, b,
                                                (short)0, acc, false, false);
  }
  float* cp = C + (long)(tm * 16 + kh * 4) * NCOLS + tn * 16 + row;
#pragma unroll
  for (int j = 0; j < 8; ++j) cp[(long)j * NCOLS] = acc[j];
}

// ---------------------------------------------------------------- edge pass 1: logits + segment max
// one wave per edge; lane covers H*C/32 contiguous channels (all in one head)

template <int H, int C>
__global__ void edge_logits_kernel(const float* __restrict__ xl,
                                   const float* __restrict__ xr,
                                   const float* __restrict__ att,
                                   const int* __restrict__ ei,
                                   float* __restrict__ logits,
                                   unsigned* __restrict__ lmaxKey) {
  const int HC  = H * C;
  const int CPL = HC / 32;   // channels per lane
  const int LPH = C / CPL;   // lanes per head
  int wave = (blockIdx.x * blockDim.x + threadIdx.x) >> 5;
  int lane = threadIdx.x & 31;
  if (wave >= E_TOT) return;
  int s, d;
  edge_nodes(wave, ei, s, d);
  const float* pl = xl + (long)s * HC + lane * CPL;
  const float* pr = xr + (long)d * HC + lane * CPL;
  const float* pa = att + lane * CPL;
  float p = 0.0f;
#pragma unroll
  for (int i = 0; i < CPL; ++i) {
    float m = pl[i] + pr[i];
    m = (m > 0.0f) ? m : 0.2f * m;   // leaky_relu, slope 0.2
    p += m * pa[i];
  }
#pragma unroll
  for (int off = 1; off < LPH; off <<= 1) p += __shfl_xor(p, off, 32);
  if ((lane % LPH) == 0) {
    int h = lane / LPH;
    logits[(long)wave * H + h] = p;
    atomicMax(&lmaxKey[d * H + h], fkey(p));
  }
}

// ---------------------------------------------------------------- edge pass 2: exp + segment sum

template <int H>
__global__ void edge_exp_kernel(float* __restrict__ logits,
                                const unsigned* __restrict__ lmaxKey,
                                float* __restrict__ denom,
                                const int* __restrict__ ei) {
  long tid = blockIdx.x * (long)blockDim.x + threadIdx.x;
  if (tid >= (long)E_TOT * H) return;
  int e = (int)(tid / H);
  int h = (int)(tid - (long)e * H);
  int s, d;
  edge_nodes(e, ei, s, d);
  float ex = __expf(logits[tid] - fdec(lmaxKey[d * H + h]));
  logits[tid] = ex;                 // overwrite logits with exp value
  atomicAdd(&denom[d * H + h], ex);
}

// ---------------------------------------------------------------- edge pass 3: alpha-weighted scatter

template <int H, int C>
__global__ void edge_aggregate_kernel(const float* __restrict__ xl,
                                      const float* __restrict__ ex,
                                      const float* __restrict__ denom,
                                      const int* __restrict__ ei,
                                      float* __restrict__ out) {
  const int HC  = H * C;
  const int CPL = HC / 32;
  int wave = (blockIdx.x * blockDim.x + threadIdx.x) >> 5;
  int lane = threadIdx.x & 31;
  if (wave >= E_TOT) return;
  int s, d;
  edge_nodes(wave, ei, s, d);
  int h = (lane * CPL) / C;
  float alpha = ex[(long)wave * H + h] / (denom[d * H + h] + 1e-16f);
  const float* pl = xl + (long)s * HC + lane * CPL;
  float* po = out + (long)d * HC + lane * CPL;
#pragma unroll
  for (int i = 0; i < CPL; ++i) atomicAdd(&po[i], alpha * pl[i]);
}

// ---------------------------------------------------------------- bias + ELU (in place)

__global__ void bias_elu_kernel(float* __restrict__ h, const float* __restrict__ b,
                                long n, int cols) {
  long t = blockIdx.x * (long)blockDim.x + threadIdx.x;
  if (t >= n) return;
  float v = h[t] + b[(int)(t % cols)];
  h[t] = (v > 0.0f) ? v : (__expf(v) - 1.0f);
}

// ---------------------------------------------------------------- final batch mean + bias

__global__ void final_mean_kernel(const float* __restrict__ out2,
                                  const float* __restrict__ b2,
                                  float* __restrict__ dout) {
  int b = blockIdx.x;     // 8
  int c = threadIdx.x;    // 128
  const float* p = out2 + (long)b * NNODE * COUT + c;
  float s = 0.0f;
  for (int n = 0; n < NNODE; ++n) s += p[(long)n * COUT];
  dout[b * COUT + c] = s * (1.0f / (float)NNODE) + b2[c];
}

// ---------------------------------------------------------------- launch

extern "C" void kernel_launch(void* const* d_in, const int* in_sizes, int n_in,
                              void* d_out, int out_size, void* d_ws, size_t ws_size,
                              hipStream_t stream) {
  const float* input  = (const float*)d_in[0];
  const int*   ei     = (const int*)d_in[1];
  const float* W_temb = (const float*)d_in[2];
  const float* b_temb = (const float*)d_in[3];
  const float* Wl1    = (const float*)d_in[4];
  const float* Wr1    = (const float*)d_in[5];
  const float* att1   = (const float*)d_in[6];
  const float* b1     = (const float*)d_in[7];
  const float* Wl2    = (const float*)d_in[8];
  const float* Wr2    = (const float*)d_in[9];
  const float* att2   = (const float*)d_in[10];
  const float* b2     = (const float*)d_in[11];
  float* out          = (float*)d_out;
  float* ws           = (float*)d_ws;

  // workspace layout (float offsets)
  const long OFF_X    = 0L;                       // x [16384,128]; reused as xl2
  const long OFF_XL1  = OFF_X   + (long)NTOT * CIN;   // xl1 [16384,256]
  const long OFF_XR1  = OFF_XL1 + (long)NTOT * HC1;   // xr1 [16384,256]; reused: xr2 + out2
  const long OFF_H    = OFF_XR1 + (long)NTOT * HC1;   // out1 / h [16384,256]
  const long OFF_LOG  = OFF_H   + (long)NTOT * HC1;   // logits/ex [E,4]
  const long OFF_LMAX = OFF_LOG + (long)E_TOT * 4;    // lmax keys [16384*4] (u32)
  const long OFF_DEN  = OFF_LMAX + (long)NTOT * 4;    // denom [16384*4]
  const long OFF_XL2  = OFF_X;
  const long OFF_XR2  = OFF_XR1;
  const long OFF_OUT2 = OFF_XR1 + (long)NTOT * COUT;

  float*    x    = ws + OFF_X;
  float*    xl1  = ws + OFF_XL1;
  float*    xr1  = ws + OFF_XR1;
  float*    hbuf = ws + OFF_H;
  float*    logi = ws + OFF_LOG;
  unsigned* lmax = (unsigned*)(ws + OFF_LMAX);
  float*    den  = ws + OFF_DEN;
  float*    xl2  = ws + OFF_XL2;
  float*    xr2  = ws + OFF_XR2;
  float*    out2 = ws + OFF_OUT2;

  auto blk = [](long threads) { return (int)((threads + TPB - 1) / TPB); };
  const long edgeThreads = (long)E_TOT * 32;   // wave per edge

  // ---- stage 0: temporal embedding
  temb_gemm_kernel<<<blk((long)(NTOT / 16) * 8 * 32), TPB, 0, stream>>>(
      input, W_temb, b_temb, x);

  // ---- layer 1
  zero_kernel<<<blk((long)NTOT * HC1), TPB, 0, stream>>>(hbuf, (long)NTOT * HC1);
  zero_kernel<<<blk((long)NTOT * 8), TPB, 0, stream>>>(ws + OFF_LMAX, (long)NTOT * 8);

  wmma_gemm_f32<CIN, HC1><<<blk((long)(NTOT / 16) * (HC1 / 16) * 32), TPB, 0, stream>>>(x, Wl1, xl1);
  wmma_gemm_f32<CIN, HC1><<<blk((long)(NTOT / 16) * (HC1 / 16) * 32), TPB, 0, stream>>>(x, Wr1, xr1);

  edge_logits_kernel<4, 64><<<blk(edgeThreads), TPB, 0, stream>>>(xl1, xr1, att1, ei, logi, lmax);
  edge_exp_kernel<4><<<blk((long)E_TOT * 4), TPB, 0, stream>>>(logi, lmax, den, ei);
  edge_aggregate_kernel<4, 64><<<blk(edgeThreads), TPB, 0, stream>>>(xl1, logi, den, ei, hbuf);

  bias_elu_kernel<<<blk((long)NTOT * HC1), TPB, 0, stream>>>(hbuf, b1, (long)NTOT * HC1, HC1);

  // ---- layer 2
  zero_kernel<<<blk((long)NTOT * COUT), TPB, 0, stream>>>(out2, (long)NTOT * COUT);
  zero_kernel<<<blk((long)NTOT * 8), TPB, 0, stream>>>(ws + OFF_LMAX, (long)NTOT * 8);

  wmma_gemm_f32<HC1, COUT><<<blk((long)(NTOT / 16) * (COUT / 16) * 32), TPB, 0, stream>>>(hbuf, Wl2, xl2);
  wmma_gemm_f32<HC1, COUT><<<blk((long)(NTOT / 16) * (COUT / 16) * 32), TPB, 0, stream>>>(hbuf, Wr2, xr2);

  edge_logits_kernel<1, 128><<<blk(edgeThreads), TPB, 0, stream>>>(xl2, xr2, att2, ei, logi, lmax);
  edge_exp_kernel<1><<<blk((long)E_TOT), TPB, 0, stream>>>(logi, lmax, den, ei);
  edge_aggregate_kernel<1, 128><<<blk(edgeThreads), TPB, 0, stream>>>(xl2, logi, den, ei, out2);

  // ---- final: per-batch node mean + bias
  final_mean_kernel<<<8, COUT, 0, stream>>>(out2, b2, out);
}